// KalmanFilterPredictor_89395449299471
// MI455X (gfx1250) — compile-verified
//
#include <hip/hip_runtime.h>
#include <hip/hip_bf16.h>

#define STATE_DIM 128
#define OBS_DIM   64
#define BATCH     2048
#define T_STEPS   128      // (1024*8)/64

typedef float v2f __attribute__((ext_vector_type(2)));
typedef float v8f __attribute__((ext_vector_type(8)));

#define WMMA_F32(a, b, c) \
    __builtin_amdgcn_wmma_f32_16x16x4_f32(false, (a), false, (b), (short)0, (c), false, false)

// ---------------------------------------------------------------------------
// Kernel 1: batch-independent Riccati precompute (serial over t), WMMA-based.
// Produces A_t = (I - K_t H) F  [128x128 row-major]  and  K_t [128x64].
// 512 threads = 16 waves (cheaper barriers, more VGPR headroom per wave);
// all big matmuls on V_WMMA_F32_16X16X4_F32 with fragments PRELOADED into
// register arrays per tile (one load clause + one wait, then a back-to-back
// WMMA accumulation chain). S^-1 via 2-barrier unnormalized Gauss-Jordan.
//
// Fragment conventions (f32 16x16x4, wave32):
//   A-frag: lane -> m = lane%16 ; regs j + half lh=lane/16 -> k = k0+2*lh+j
//   B-frag: lane -> n = lane%16 ; regs/half -> k  (Bfrag[k][n] = src[n][k]
//           reads row-major "transposed" operands contiguously)
//   D:      lane -> n = lane%16 ; reg v -> m = v + 8*lh
// ---------------------------------------------------------------------------
__global__ __launch_bounds__(512)
void kf_precompute(const float* __restrict__ F, const float* __restrict__ H,
                   const float* __restrict__ Q, const float* __restrict__ R,
                   const float* __restrict__ P0,
                   float* __restrict__ A_all,   // [T,128,128]
                   float* __restrict__ K_all,   // [T,128,64]
                   float* __restrict__ scr)     // scratch: T1,Pm,Pp,T2,HF
{
    const int tid  = threadIdx.x;
    const int NT   = 512;
    const int lane = tid & 31;
    const int wave = tid >> 5;     // 0..15
    const int lm   = lane & 15;    // m (A-frag) / n (B-frag, D)
    const int lh   = lane >> 4;    // k-half selector

    float* T1 = scr;               // [128,128]
    float* Pm = scr + 16384;       // [128,128]
    float* Pp = scr + 32768;       // [128,128]
    float* T2 = scr + 49152;       // [64,128]  = H @ Pm
    float* HF = scr + 57344;       // [64,128]  = H @ F  (time-invariant)

    __shared__ float Saug[64 * 129];   // [S | I] augmented, padded rows
    __shared__ float fac[64];

    // ---- init: Pp = P0 copy; HF = H @ F (scalar, once) ----
    for (int i = tid; i < 128 * 128; i += NT) Pp[i] = P0[i];
    for (int idx = tid; idx < 64 * 128; idx += NT) {
        int i = idx >> 7, j = idx & 127;
        float s = 0.f;
        for (int k = 0; k < 128; ++k) s += H[i * 128 + k] * F[k * 128 + j];
        HF[idx] = s;
    }
    __syncthreads();

    for (int t = 0; t < T_STEPS; ++t) {
        // ---- T1 = F @ Pp  (64 tiles, 4 per wave) ----
        for (int tt = wave; tt < 64; tt += 16) {
            int mt = tt >> 3, nt = tt & 7;
            const float* Ab = F  + (mt * 16 + lm) * 128 + 2 * lh;   // contiguous
            const float* Bb = Pp + (2 * lh) * 128 + nt * 16 + lm;   // strided cols
            v2f af[32], bf[32];
            #pragma unroll
            for (int j = 0; j < 32; ++j) af[j] = *(const v2f*)(Ab + 4 * j);
            #pragma unroll
            for (int j = 0; j < 32; ++j) {
                bf[j].x = Bb[(4 * j) * 128];
                bf[j].y = Bb[(4 * j + 1) * 128];
            }
            v8f acc = {};
            #pragma unroll
            for (int j = 0; j < 32; ++j) acc = WMMA_F32(af[j], bf[j], acc);
            #pragma unroll
            for (int v = 0; v < 8; ++v)
                T1[(mt * 16 + v + 8 * lh) * 128 + nt * 16 + lm] = acc[v];
        }
        __syncthreads();

        // ---- Pm = T1 @ F^T + Q  (B = F^T -> contiguous rows of F) ----
        for (int tt = wave; tt < 64; tt += 16) {
            int mt = tt >> 3, nt = tt & 7;
            const float* Ab = T1 + (mt * 16 + lm) * 128 + 2 * lh;
            const float* Bb = F  + (nt * 16 + lm) * 128 + 2 * lh;
            v2f af[32], bf[32];
            #pragma unroll
            for (int j = 0; j < 32; ++j) af[j] = *(const v2f*)(Ab + 4 * j);
            #pragma unroll
            for (int j = 0; j < 32; ++j) bf[j] = *(const v2f*)(Bb + 4 * j);
            v8f acc;
            #pragma unroll
            for (int v = 0; v < 8; ++v)
                acc[v] = Q[(mt * 16 + v + 8 * lh) * 128 + nt * 16 + lm];
            #pragma unroll
            for (int j = 0; j < 32; ++j) acc = WMMA_F32(af[j], bf[j], acc);
            #pragma unroll
            for (int v = 0; v < 8; ++v)
                Pm[(mt * 16 + v + 8 * lh) * 128 + nt * 16 + lm] = acc[v];
        }
        __syncthreads();

        // ---- T2 = H @ Pm  (32 tiles, 2 per wave) ----
        for (int tt = wave; tt < 32; tt += 16) {
            int mt = tt >> 3, nt = tt & 7;
            const float* Ab = H  + (mt * 16 + lm) * 128 + 2 * lh;
            const float* Bb = Pm + (2 * lh) * 128 + nt * 16 + lm;
            v2f af[32], bf[32];
            #pragma unroll
            for (int j = 0; j < 32; ++j) af[j] = *(const v2f*)(Ab + 4 * j);
            #pragma unroll
            for (int j = 0; j < 32; ++j) {
                bf[j].x = Bb[(4 * j) * 128];
                bf[j].y = Bb[(4 * j + 1) * 128];
            }
            v8f acc = {};
            #pragma unroll
            for (int j = 0; j < 32; ++j) acc = WMMA_F32(af[j], bf[j], acc);
            #pragma unroll
            for (int v = 0; v < 8; ++v)
                T2[(mt * 16 + v + 8 * lh) * 128 + nt * 16 + lm] = acc[v];
        }
        // right half of Saug := I (disjoint LDS columns from the S tiles below)
        for (int idx = tid; idx < 64 * 64; idx += NT) {
            int i = idx >> 6, j = idx & 63;
            Saug[i * 129 + 64 + j] = (i == j) ? 1.f : 0.f;
        }
        __syncthreads();

        // ---- Saug[:, :64] = S = T2 @ H^T + R  (16 tiles, 1 per wave) ----
        {
            int tt = wave;
            int mt = tt >> 2, nt = tt & 3;
            const float* Ab = T2 + (mt * 16 + lm) * 128 + 2 * lh;
            const float* Bb = H  + (nt * 16 + lm) * 128 + 2 * lh;   // B = H^T
            v2f af[32], bf[32];
            #pragma unroll
            for (int j = 0; j < 32; ++j) af[j] = *(const v2f*)(Ab + 4 * j);
            #pragma unroll
            for (int j = 0; j < 32; ++j) bf[j] = *(const v2f*)(Bb + 4 * j);
            v8f acc;
            #pragma unroll
            for (int v = 0; v < 8; ++v)
                acc[v] = R[(mt * 16 + v + 8 * lh) * 64 + nt * 16 + lm];
            #pragma unroll
            for (int j = 0; j < 32; ++j) acc = WMMA_F32(af[j], bf[j], acc);
            #pragma unroll
            for (int v = 0; v < 8; ++v)
                Saug[(mt * 16 + v + 8 * lh) * 129 + nt * 16 + lm] = acc[v];
        }
        __syncthreads();

        // ---- Gauss-Jordan, unnormalized (2 barriers/pivot) ----
        for (int p = 0; p < 64; ++p) {
            if (tid < 64)
                fac[tid] = Saug[tid * 129 + p] * (1.0f / Saug[p * 129 + p]);
            __syncthreads();
            // eliminate rows r != p against the UNSCALED pivot row (row p
            // is never written here -> no read/write race)
            for (int idx = tid; idx < 64 * 128; idx += NT) {
                int r = idx >> 7, c = idx & 127;
                if (r != p) Saug[r * 129 + c] -= fac[r] * Saug[p * 129 + c];
            }
            __syncthreads();
        }
        // left half is now diag(d); S^-1 = diag(1/d) * right half
        if (tid < 64) fac[tid] = 1.0f / Saug[tid * 129 + tid];
        __syncthreads();
        for (int idx = tid; idx < 64 * 64; idx += NT) {
            int i = idx >> 6, j = idx & 63;
            Saug[i * 129 + 64 + j] *= fac[i];
        }
        __syncthreads();

        // ---- K = T2^T @ S^-1  (Pm H^T = T2^T by symmetry; 32 tiles) ----
        float* Kt = K_all + (size_t)t * 128 * 64;
        for (int tt = wave; tt < 32; tt += 16) {
            int mt = tt >> 2, nt = tt & 3;
            const float* Ab = T2 + (2 * lh) * 128 + mt * 16 + lm;   // A = T2^T
            const float* Bb = &Saug[(2 * lh) * 129 + 64 + nt * 16 + lm];
            v2f af[16], bf[16];
            #pragma unroll
            for (int j = 0; j < 16; ++j) {
                af[j].x = Ab[(4 * j) * 128];
                af[j].y = Ab[(4 * j + 1) * 128];
            }
            #pragma unroll
            for (int j = 0; j < 16; ++j) {
                bf[j].x = Bb[(4 * j) * 129];
                bf[j].y = Bb[(4 * j + 1) * 129];
            }
            v8f acc = {};
            #pragma unroll
            for (int j = 0; j < 16; ++j) acc = WMMA_F32(af[j], bf[j], acc);
            #pragma unroll
            for (int v = 0; v < 8; ++v)
                Kt[(mt * 16 + v + 8 * lh) * 64 + nt * 16 + lm] = acc[v];
        }
        __syncthreads();

        // ---- fused: Pp = Pm - K @ T2  and  A_t = F - K @ HF ----
        // (shared negated-K A-fragments; f32 WMMA has no A/B negation, only CNeg)
        float* At = A_all + (size_t)t * 128 * 128;
        for (int tt = wave; tt < 64; tt += 16) {
            int mt = tt >> 3, nt = tt & 7;
            const float* Ab  = Kt + (mt * 16 + lm) * 64 + 2 * lh;
            const float* Bb1 = T2 + (2 * lh) * 128 + nt * 16 + lm;
            const float* Bb2 = HF + (2 * lh) * 128 + nt * 16 + lm;
            v2f af[16], b1[16], b2[16];
            #pragma unroll
            for (int j = 0; j < 16; ++j) af[j] = -*(const v2f*)(Ab + 4 * j);
            #pragma unroll
            for (int j = 0; j < 16; ++j) {
                b1[j].x = Bb1[(4 * j) * 128];
                b1[j].y = Bb1[(4 * j + 1) * 128];
            }
            #pragma unroll
            for (int j = 0; j < 16; ++j) {
                b2[j].x = Bb2[(4 * j) * 128];
                b2[j].y = Bb2[(4 * j + 1) * 128];
            }
            v8f accP, accA;
            #pragma unroll
            for (int v = 0; v < 8; ++v) {
                int row = (mt * 16 + v + 8 * lh) * 128 + nt * 16 + lm;
                accP[v] = Pm[row];
                accA[v] = F[row];
            }
            #pragma unroll
            for (int j = 0; j < 16; ++j) {
                accP = WMMA_F32(af[j], b1[j], accP);
                accA = WMMA_F32(af[j], b2[j], accA);
            }
            #pragma unroll
            for (int v = 0; v < 8; ++v) {
                int row = (mt * 16 + v + 8 * lh) * 128 + nt * 16 + lm;
                Pp[row] = accP[v];
                At[row] = accA[v];
            }
        }
        __syncthreads();
    }
}

// ---------------------------------------------------------------------------
// Kernel 2: batched linear recurrence  x_t = A_t x_{t-1} + K_t z_t
// via V_WMMA_F32_16X16X4_F32. Each wave owns 16 samples; state tile kept in
// a private LDS region (stride 130 floats -> conflict-free b64 frag loads).
// Bfrag[k][n] = A_t[nG][kG] reads row-major A_t directly (no transpose).
// ---------------------------------------------------------------------------
__global__ __launch_bounds__(128)
void kf_batch(const float* __restrict__ feats,   // [2048, 8192]  (z_t row = s*8192 + t*64)
              const float* __restrict__ x0,      // [128]
              const float* __restrict__ A_all,   // [T,128,128]
              const float* __restrict__ K_all,   // [T,128,64]
              float* __restrict__ out)           // [2048, T, 128]
{
    constexpr int WAVES   = 4;
    constexpr int XSTRIDE = 130;                 // padded: banks (2m+k)%64, conflict-free
    __shared__ float Xs[WAVES * 16 * XSTRIDE];

    const int lane  = threadIdx.x & 31;
    const int wave  = threadIdx.x >> 5;
    const int lm    = lane & 15;                 // m (A-frag) or n (B-frag)
    const int lh    = lane >> 4;                 // k-half selector
    const int sBase = blockIdx.x * (WAVES * 16) + wave * 16;

    float* X = &Xs[wave * 16 * XSTRIDE];         // private 16x128 state tile

    // init per-sample state: X[m][k] = x0[k]
    for (int i = lane; i < 16 * STATE_DIM; i += 32) {
        int m = i >> 7, k = i & 127;
        X[m * XSTRIDE + k] = x0[k];
    }
    // only this wave touches X; per-wave LDS ops are in-order, no barrier needed

    #pragma unroll 1
    for (int t = 0; t < T_STEPS; ++t) {
        const float* At = A_all + (size_t)t * 128 * 128;
        const float* Kt = K_all + (size_t)t * 128 * 64;

        // z_t A-fragments: 16 k-steps over OBS_DIM=64
        v2f zf[16];
        const float* zrow = feats + (size_t)(sBase + lm) * 8192 + t * 64 + 2 * lh;
        #pragma unroll
        for (int j = 0; j < 16; ++j)
            zf[j] = *(const v2f*)(zrow + j * 4);

        // state A-fragments: 32 k-steps over STATE_DIM=128 (from LDS)
        v2f xf[32];
        const float* xrow = &X[lm * XSTRIDE + 2 * lh];
        #pragma unroll
        for (int j = 0; j < 32; ++j)
            xf[j] = *(const v2f*)(xrow + j * 4);

        for (int nt = 0; nt < 8; ++nt) {
            v8f acc = {};
            const float* Ab = At + (nt * 16 + lm) * 128 + 2 * lh;
            #pragma unroll
            for (int j = 0; j < 32; ++j) {
                v2f bf = *(const v2f*)(Ab + j * 4);
                acc = WMMA_F32(xf[j], bf, acc);
            }
            const float* Kb = Kt + (nt * 16 + lm) * 64 + 2 * lh;
            #pragma unroll
            for (int j = 0; j < 16; ++j) {
                v2f bf = *(const v2f*)(Kb + j * 4);
                acc = WMMA_F32(zf[j], bf, acc);
            }
            // D: lane holds (m = v + 8*lh, n = lm); write out + refresh state tile
            #pragma unroll
            for (int v = 0; v < 8; ++v) {
                int m = v + 8 * lh;
                float val = acc[v];
                out[(size_t)(sBase + m) * (T_STEPS * STATE_DIM) + t * STATE_DIM + nt * 16 + lm] = val;
                X[m * XSTRIDE + nt * 16 + lm] = val;
            }
        }
    }
}

extern "C" void kernel_launch(void* const* d_in, const int* in_sizes, int n_in,
                              void* d_out, int out_size, void* d_ws, size_t ws_size,
                              hipStream_t stream) {
    const float* feats = (const float*)d_in[0];
    const float* F     = (const float*)d_in[1];
    const float* H     = (const float*)d_in[2];
    const float* Q     = (const float*)d_in[3];
    const float* R     = (const float*)d_in[4];
    const float* P     = (const float*)d_in[5];
    const float* x     = (const float*)d_in[6];
    float* out = (float*)d_out;
    float* ws  = (float*)d_ws;

    float* A_all = ws;                                   // 128*128*128
    float* K_all = A_all + (size_t)T_STEPS * 128 * 128;  // 128*128*64
    float* scr   = K_all + (size_t)T_STEPS * 128 * 64;   // 65536 floats

    kf_precompute<<<1, 512, 0, stream>>>(F, H, Q, R, P, A_all, K_all, scr);
    kf_batch<<<BATCH / 64, 128, 0, stream>>>(feats, x, A_all, K_all, out);
}